// DynamicDistanceMessagePassing_30477087933113
// MI455X (gfx1250) — compile-verified
//
#include <hip/hip_runtime.h>

#define VN    100000
#define KN    32
#define FINC  128     // input feature width == f_in of every layer's GEMM
#define NF    64      // per-layer output features (before mean/max concat)
#define LDOUT 640     // output row stride: 4*128 + 128
#define WSTR  160     // LDS dword stride per K-pair row; 160 % 64 == 32 -> conflict-free

typedef __attribute__((ext_vector_type(2))) float v2f;
typedef __attribute__((ext_vector_type(8))) float v8f;

// ---------------------------------------------------------------- copy x tail
__global__ void copy_x_kernel(const float* __restrict__ x, float* __restrict__ out) {
    int i = blockIdx.x * blockDim.x + threadIdx.x;
    if (i >= VN * FINC) return;
    int v = i / FINC, c = i % FINC;
    out[(size_t)v * LDOUT + 4 * FINC + c] = x[i];
}

// ------------------------------------------------- d *= 10*sigmoid(din @ dw + db)
// one wave per vertex; lane k (K==32==wave32) scales d[v][k]
__global__ void dist_scale_kernel(const float* __restrict__ x,
                                  const float* __restrict__ fpart,   // nullptr on layer 0; row stride LDOUT
                                  const float* __restrict__ dw,
                                  const float* __restrict__ db,
                                  float* __restrict__ d_run) {
    int wave = threadIdx.x >> 5, lane = threadIdx.x & 31;
    int v = blockIdx.x * 8 + wave;
    if (v >= VN) return;
    const float* xr = x + (size_t)v * FINC;
    float s = 0.f;
    #pragma unroll
    for (int c = 0; c < FINC; c += 32) s += xr[c + lane] * dw[c + lane];
    if (fpart) {
        const float* fr = fpart + (size_t)v * LDOUT;
        #pragma unroll
        for (int c = 0; c < FINC; c += 32) s += fr[c + lane] * dw[FINC + c + lane];
    }
    #pragma unroll
    for (int off = 16; off > 0; off >>= 1) s += __shfl_xor(s, off, 32);
    s += db[0];
    float scale = 10.f / (1.f + __expf(-s));
    d_run[(size_t)v * KN + lane] *= scale;
}

// ------------------------------------------- feat_out = relu(A[V,128] @ W[128,64] + b)
// 8 waves/block; each wave owns a 16-row x 64-col tile via V_WMMA_F32_16X16X4_F32.
// Weights staged in LDS K-pair interleaved: sW2[p=k/2][n][r=k%2] so each B v2f
// fragment is one contiguous ds_load_b64 (no fragment-assembly movs, no conflicts).
__global__ void feat_gemm_kernel(const float* __restrict__ A, int ldA,
                                 const float* __restrict__ W,     // [128,64] row-major
                                 const float* __restrict__ bias,  // [64]
                                 float* __restrict__ out) {       // [VN,64]
    __shared__ float sW2[(FINC / 2) * WSTR];         // 64 * 160 dwords = 40 KB
    for (int i = threadIdx.x; i < FINC * NF; i += blockDim.x) {
        int kk = i / NF, nn = i % NF;
        sW2[(kk >> 1) * WSTR + nn * 2 + (kk & 1)] = W[i];
    }
    __syncthreads();

    int wave = threadIdx.x >> 5, lane = threadIdx.x & 31;
    int row0 = (blockIdx.x * 8 + wave) * 16;
    if (row0 >= VN) return;                          // wave-uniform: EXEC all-ones below

    int half = lane >> 4;                            // 0: lanes 0-15, 1: lanes 16-31
    int l15  = lane & 15;

    v8f acc0 = {}, acc1 = {}, acc2 = {}, acc3 = {};
    // A fragment: lanes 0-15 hold (M=l15, K=k,k+1); lanes 16-31 hold (M=l15, K=k+2,k+3)
    const float* arow = A + (size_t)(row0 + l15) * ldA + half * 2;
    // B fragments: lo half reads K-pair p=k/2, hi half p=k/2+1; one b64 per nt tile
    const float* bbase = sW2 + (size_t)half * WSTR + l15 * 2;

    #pragma unroll 8
    for (int k = 0; k < FINC; k += 4) {
        v2f a;
        a.x = arow[k];
        a.y = arow[k + 1];
        const float* bp = bbase + (k >> 1) * WSTR;
        v2f b0 = *(const v2f*)(bp);
        v2f b1 = *(const v2f*)(bp + 32);
        v2f b2 = *(const v2f*)(bp + 64);
        v2f b3 = *(const v2f*)(bp + 96);
        acc0 = __builtin_amdgcn_wmma_f32_16x16x4_f32(false, a, false, b0, (short)0, acc0, false, false);
        acc1 = __builtin_amdgcn_wmma_f32_16x16x4_f32(false, a, false, b1, (short)0, acc1, false, false);
        acc2 = __builtin_amdgcn_wmma_f32_16x16x4_f32(false, a, false, b2, (short)0, acc2, false, false);
        acc3 = __builtin_amdgcn_wmma_f32_16x16x4_f32(false, a, false, b3, (short)0, acc3, false, false);
    }

    // D layout: VGPR r -> (M=r, N=l15) for lanes 0-15, (M=r+8, N=l15) for lanes 16-31
    int m0 = row0 + half * 8;
    #pragma unroll
    for (int r = 0; r < 8; ++r) {
        float* orow = out + (size_t)(m0 + r) * NF;
        float t;
        t = acc0[r] + bias[l15];      orow[l15]      = t > 0.f ? t : 0.f;
        t = acc1[r] + bias[16 + l15]; orow[16 + l15] = t > 0.f ? t : 0.f;
        t = acc2[r] + bias[32 + l15]; orow[32 + l15] = t > 0.f ? t : 0.f;
        t = acc3[r] + bias[48 + l15]; orow[48 + l15] = t > 0.f ? t : 0.f;
    }
}

// ---------------------------- accumulate_knn: mean/max over K of exp(-10 d)*feat[idx]
// one wave per vertex; lane owns 2 feature columns; lane k owns weight w[k]
__global__ void accum_kernel(const float* __restrict__ d_run,
                             const int*   __restrict__ idx,
                             const float* __restrict__ feat,   // [VN,64] (post-relu prev)
                             float* __restrict__ out, int colOff) {
    int wave = threadIdx.x >> 5, lane = threadIdx.x & 31;
    int v = blockIdx.x * 8 + wave;
    if (v >= VN) return;
    float w = __expf(-10.f * d_run[(size_t)v * KN + lane]);
    const int* nb = idx + (size_t)v * KN;
    int c0 = lane * 2;
    float m0 = 0.f, m1 = 0.f;
    float x0 = -__builtin_inff(), x1 = -__builtin_inff();
    #pragma unroll 8
    for (int k = 0; k < KN; ++k) {
        float wk = __shfl(w, k, 32);
        int j = nb[k];
        const float* fr = feat + (size_t)j * NF + c0;
        float f0 = fr[0] * wk, f1 = fr[1] * wk;
        m0 += f0; m1 += f1;
        x0 = fmaxf(x0, f0);
        x1 = fmaxf(x1, f1);
    }
    m0 *= (1.f / KN); m1 *= (1.f / KN);
    const float* pr = feat + (size_t)v * NF + c0;
    float p0 = pr[0], p1 = pr[1];
    float* orow = out + (size_t)v * LDOUT + colOff;
    orow[c0]          = m0 - p0;
    orow[c0 + 1]      = m1 - p1;
    orow[NF + c0]     = x0 - p0;
    orow[NF + c0 + 1] = x1 - p1;
}

extern "C" void kernel_launch(void* const* d_in, const int* in_sizes, int n_in,
                              void* d_out, int out_size, void* d_ws, size_t ws_size,
                              hipStream_t stream) {
    const float* x   = (const float*)d_in[0];
    const int*   idx = (const int*)d_in[1];
    const float* dsq = (const float*)d_in[2];
    const float *dw[4], *db[4], *fw[4], *fb[4];
    for (int i = 0; i < 4; ++i) {
        dw[i] = (const float*)d_in[3 + 4 * i + 0];
        db[i] = (const float*)d_in[3 + 4 * i + 1];
        fw[i] = (const float*)d_in[3 + 4 * i + 2];
        fb[i] = (const float*)d_in[3 + 4 * i + 3];
    }
    float* out   = (float*)d_out;
    float* d_run = (float*)d_ws;                     // [VN,KN] running rescaled distances
    float* prev  = d_run + (size_t)VN * KN;          // [VN,NF] per-layer relu features

    // fresh running distances every call (input is never mutated)
    hipMemcpyAsync(d_run, dsq, (size_t)VN * KN * sizeof(float),
                   hipMemcpyDeviceToDevice, stream);

    copy_x_kernel<<<(VN * FINC + 255) / 256, 256, 0, stream>>>(x, out);

    const int pvBlocks   = (VN + 7) / 8;             // wave-per-vertex kernels
    const int gemmBlocks = (VN / 16 + 7) / 8;        // 16-row tiles, 8 waves/block

    for (int i = 0; i < 4; ++i) {
        const float* fpart = (i == 0) ? nullptr : out + (size_t)(i - 1) * FINC;
        dist_scale_kernel<<<pvBlocks, 256, 0, stream>>>(x, fpart, dw[i], db[i], d_run);

        const float* A = (i == 0) ? x : out + (size_t)(i - 1) * FINC;
        int ldA        = (i == 0) ? FINC : LDOUT;
        feat_gemm_kernel<<<gemmBlocks, 256, 0, stream>>>(A, ldA, fw[i], fb[i], prev);

        accum_kernel<<<pvBlocks, 256, 0, stream>>>(d_run, idx, prev, out, i * FINC);
    }
}